// SingleIterCCPOptimizer_57982058496075
// MI455X (gfx1250) — compile-verified
//
#include <hip/hip_runtime.h>
#include <math.h>

#define S_DIM 1024
#define K_DIM 4

typedef float v2f __attribute__((ext_vector_type(2)));
typedef float v8f __attribute__((ext_vector_type(8)));

// ---- workspace layout (bytes) ----
// [0   .. 127]  float ws_f[32]: 0 lpS, 1 lpD, 2 lpT, 3 pS, 4 pD, 5 pT, 6 pL,
//                               8..11 lq[k], 16..23 m[8] (A rows then B rows)
// [128 .. 255]  int   ws_i[32]: 0 root, 1 leaf, 2..5 cA[k], 6..9 cB[k]
// [256 ..]      float X[S_DIM*16]   (row j holds K-slice j, 16 columns)
// [256+64K ..]  float Y[S_DIM*16]   (Y = R @ X)

__device__ __forceinline__ float softplusf(float x) {
    return fmaxf(x, 0.f) + log1pf(expf(-fabsf(x)));
}
__device__ __forceinline__ float laddexp(float a, float b) {
    float mx = fmaxf(a, b), mn = fminf(a, b);
    return mx + log1pf(expf(mn - mx));
}
__device__ __forceinline__ float lse4(float a, float b, float c, float d) {
    float mx = fmaxf(fmaxf(a, b), fmaxf(c, d));
    return mx + logf(expf(a - mx) + expf(b - mx) + expf(c - mx) + expf(d - mx));
}

// ---------------- Phase 1: scalars + row maxima -----------------------------
__global__ void k_prep(const float* __restrict__ log_delta,
                       const float* __restrict__ log_tau,
                       const float* __restrict__ log_lambda,
                       const float* __restrict__ log_Pi,
                       const int*   __restrict__ splits_c1,
                       const int*   __restrict__ splits_c2,
                       const float* __restrict__ log_q,
                       const unsigned char* __restrict__ is_leaf,
                       const int*   __restrict__ root_id,
                       float* __restrict__ ws_f, int* __restrict__ ws_i)
{
    __shared__ int   sIdx[8];
    __shared__ float red[256];
    int t = threadIdx.x;
    if (t == 0) {
        float d  = softplusf(log_delta[0]);
        float ta = softplusf(log_tau[0]);
        float la = softplusf(log_lambda[0]);
        float rs = 1.f + d + ta + la;
        float pS = 1.f / rs, pD = d / rs, pT = ta / rs, pL = la / rs;
        ws_f[0] = logf(pS); ws_f[1] = logf(pD); ws_f[2] = logf(pT);
        ws_f[3] = pS; ws_f[4] = pD; ws_f[5] = pT; ws_f[6] = pL;
        int root = root_id[0];
        ws_i[0] = root;
        ws_i[1] = is_leaf[root] ? 1 : 0;
        for (int k = 0; k < K_DIM; ++k) {
            int ca = splits_c1[root * K_DIM + k];
            int cb = splits_c2[root * K_DIM + k];
            ws_i[2 + k] = ca; ws_i[6 + k] = cb;
            sIdx[k] = ca; sIdx[4 + k] = cb;
            ws_f[8 + k] = log_q[root * K_DIM + k];
        }
    }
    __syncthreads();
    for (int k = 0; k < 8; ++k) {
        const float* p = log_Pi + (long)sIdx[k] * S_DIM;
        float mx = -INFINITY;
        for (int j = t; j < S_DIM; j += 256) mx = fmaxf(mx, p[j]);
        red[t] = mx;
        __syncthreads();
        for (int off = 128; off > 0; off >>= 1) {
            if (t < off) red[t] = fmaxf(red[t], red[t + off]);
            __syncthreads();
        }
        if (t == 0) ws_f[16 + k] = red[0];
        __syncthreads();
    }
}

// ---------------- Phase 2: build X (K x 16 operand) -------------------------
__global__ void k_buildX(const float* __restrict__ log_Pi,
                         const float* __restrict__ E,
                         const float* __restrict__ ws_f,
                         const int*   __restrict__ ws_i,
                         float* __restrict__ X)
{
    int j = blockIdx.x * blockDim.x + threadIdx.x;
    if (j >= S_DIM) return;
    float out[16];
#pragma unroll
    for (int k = 0; k < 4; ++k) {
        out[k]     = expf(log_Pi[(long)ws_i[2 + k] * S_DIM + j] - ws_f[16 + k]);
        out[4 + k] = expf(log_Pi[(long)ws_i[6 + k] * S_DIM + j] - ws_f[20 + k]);
    }
    out[8] = E[j];
    out[9] = 1.f;                    // ones column -> rowsum(R) = cnt
#pragma unroll
    for (int c = 10; c < 16; ++c) out[c] = 0.f;
#pragma unroll
    for (int c = 0; c < 16; ++c) X[j * 16 + c] = out[c];
}

// ---------------- Phase 3: Y = R @ X via fp32 WMMA --------------------------
// One wave per 16-row M-tile; 8 waves/block, 8 blocks cover M=1024.
// A tile (16x4 f32): lanes 0-15 rows M=0..15 holding K=0,1; lanes 16-31 K=2,3.
// B tile (4x16 f32): lanes 0-15 cols N=0..15 holding K=0,1; lanes 16-31 K=2,3.
// C/D (16x16 f32): VGPR r: lanes 0-15 -> M=r, lanes 16-31 -> M=r+8.
__global__ void __launch_bounds__(256)
k_gemm(const float* __restrict__ R, const float* __restrict__ X,
       float* __restrict__ Y)
{
    __shared__ float sX[S_DIM * 16];           // 64 KB
    for (int i = threadIdx.x; i < S_DIM * 16; i += 256) sX[i] = X[i];
    __syncthreads();

    int wave  = threadIdx.x >> 5;
    int lane  = threadIdx.x & 31;
    int tile  = blockIdx.x * 8 + wave;
    int row0  = tile * 16;
    int laneM = lane & 15;
    int kо    = (lane >> 4) << 1;              // 0 for lanes 0-15, 2 for 16-31
    int ko    = kо;

    const float* aptr = R + (long)(row0 + laneM) * S_DIM + ko;
    v8f acc = {};

    for (int k0 = 0; k0 < S_DIM; k0 += 64) {
        __builtin_prefetch(aptr + k0 + 64, 0, 0);   // global_prefetch_b8
#pragma unroll
        for (int kk = 0; kk < 64; kk += 4) {
            v2f a = *(const v2f*)(aptr + k0 + kk);  // R[row][k0+kk+ko .. +1]
            v2f b;
            b.x = sX[(k0 + kk + ko) * 16 + laneM];
            b.y = sX[(k0 + kk + ko + 1) * 16 + laneM];
            acc = __builtin_amdgcn_wmma_f32_16x16x4_f32(
                false, a, false, b, (short)0, acc, false, false);
        }
    }

    int mbase = row0 + ((lane >> 4) << 3);
#pragma unroll
    for (int r = 0; r < 8; ++r)
        Y[(long)(mbase + r) * 16 + laneM] = acc[r];
}

// ---------------- Phase 4: fused log-space finalize + logsumexp -------------
__global__ void __launch_bounds__(1024)
k_final(const float* __restrict__ log_Pi, const float* __restrict__ E,
        const int* __restrict__ sC1, const int* __restrict__ sC2,
        const float* __restrict__ ws_f, const int* __restrict__ ws_i,
        const float* __restrict__ Y, float* __restrict__ out)
{
    __shared__ float sLogE[S_DIM];
    __shared__ float red[S_DIM];
    int i = threadIdx.x;

    float lpS = ws_f[0], lpD = ws_f[1], lpT = ws_f[2];
    float pS = ws_f[3], pD = ws_f[4], pT = ws_f[5], pL = ws_f[6];
    int root = ws_i[0], leaf = ws_i[1];

    float cnt  = fmaxf(Y[i * 16 + 9], 1.f);
    float Ebar = Y[i * 16 + 8] / cnt;
    int s1 = sC1[i], s2 = sC2[i];
    float Ei = E[i];
    float En = pL + pD * Ei * Ei + pT * Ei * Ebar + pS * E[s1] * E[s2];
    sLogE[i] = logf(En);
    __syncthreads();

    float lcnt = logf(cnt);
    float sk[4], dk[4], tk[4];
#pragma unroll
    for (int k = 0; k < 4; ++k) {
        long rA = (long)ws_i[2 + k] * S_DIM;
        long rB = (long)ws_i[6 + k] * S_DIM;
        float lq  = ws_f[8 + k];
        float Aki = log_Pi[rA + i], Bki = log_Pi[rB + i];
        float Af  = log_Pi[rA + s1], Ag = log_Pi[rA + s2];
        float Bf  = log_Pi[rB + s1], Bg = log_Pi[rB + s2];
        float ltA = logf(fmaxf(Y[i * 16 + k],     1e-30f)) - lcnt + ws_f[16 + k];
        float ltB = logf(fmaxf(Y[i * 16 + 4 + k], 1e-30f)) - lcnt + ws_f[20 + k];
        sk[k] = lq + laddexp(Af + Bg, Ag + Bf);
        dk[k] = lq + Aki + Bki;
        tk[k] = lq + laddexp(Aki + ltB, Bki + ltA);
    }
    float spec  = lpS + lse4(sk[0], sk[1], sk[2], sk[3]);
    float dup   = lpD + lse4(dk[0], dk[1], dk[2], dk[3]);
    float trans = lpT + lse4(tk[0], tk[1], tk[2], tk[3]);
    float sl    = lpS + laddexp(log_Pi[(long)root * S_DIM + s1] + sLogE[s2],
                                log_Pi[(long)root * S_DIM + s2] + sLogE[s1]);
    float comb = lse4(spec, dup, trans, sl);
    if (leaf) comb = log_Pi[(long)root * S_DIM + i];

    // block-wide logsumexp over 1024 lanes
    red[i] = comb;
    __syncthreads();
    for (int off = 512; off > 0; off >>= 1) {
        if (i < off) red[i] = fmaxf(red[i], red[i + off]);
        __syncthreads();
    }
    float gmax = red[0];
    __syncthreads();
    red[i] = expf(comb - gmax);
    __syncthreads();
    for (int off = 512; off > 0; off >>= 1) {
        if (i < off) red[i] += red[i + off];
        __syncthreads();
    }
    if (i == 0) out[0] = gmax + logf(red[0]);
}

extern "C" void kernel_launch(void* const* d_in, const int* in_sizes, int n_in,
                              void* d_out, int out_size, void* d_ws, size_t ws_size,
                              hipStream_t stream)
{
    const float* log_delta = (const float*)d_in[0];
    const float* log_tau   = (const float*)d_in[1];
    const float* log_lam   = (const float*)d_in[2];
    const float* E         = (const float*)d_in[3];
    const float* log_Pi    = (const float*)d_in[4];
    const int*   sC1       = (const int*)d_in[5];
    const int*   sC2       = (const int*)d_in[6];
    const float* R         = (const float*)d_in[7];
    const int*   spl1      = (const int*)d_in[8];
    const int*   spl2      = (const int*)d_in[9];
    const float* log_q     = (const float*)d_in[10];
    const unsigned char* is_leaf = (const unsigned char*)d_in[11];
    const int*   root_id   = (const int*)d_in[12];
    float* out = (float*)d_out;

    float* ws_f = (float*)d_ws;
    int*   ws_i = (int*)((char*)d_ws + 128);
    float* X    = (float*)((char*)d_ws + 256);
    float* Y    = X + S_DIM * 16;

    k_prep<<<1, 256, 0, stream>>>(log_delta, log_tau, log_lam, log_Pi,
                                  spl1, spl2, log_q, is_leaf, root_id,
                                  ws_f, ws_i);
    k_buildX<<<S_DIM / 256, 256, 0, stream>>>(log_Pi, E, ws_f, ws_i, X);
    k_gemm<<<8, 256, 0, stream>>>(R, X, Y);
    k_final<<<1, 1024, 0, stream>>>(log_Pi, E, sC1, sC2, ws_f, ws_i, Y, out);
}